// NMD_38611755991295
// MI455X (gfx1250) — compile-verified
//
#include <hip/hip_runtime.h>

typedef __attribute__((ext_vector_type(2))) float v2f;
typedef __attribute__((ext_vector_type(8))) float v8f;

#define NPTS   8192
#define NBATCH 4
#define QTILES (NPTS / 16)   // 512 query tiles per batch

// One wave (32 lanes) handles 16 query points.
// Orientation: A (16x4, M=sources) = (sx, sy, sz, |s|^2)
//              B (4x16, N=queries) = (-2qx, -2qy, -2qz, 1)
// => C[src][qry] = |s|^2 - 2 q.s ; d2 = qn(query) + C.
// C layout (wave32): vgpr r, lane ln -> (M = r + 8*(ln>=16), N = ln&15),
// so each lane packs an 8-bit source-match mask for its own query with pure
// per-lane VALU; one shfl_xor(16) merges the two source halves.
__global__ __launch_bounds__(256) void ball_query_wmma_kernel(
    const float* __restrict__ p,  // [B, N, 6] -- xyz in first 3 channels
    int* __restrict__ out)        // [B, N, 1] int32
{
    const int wavesPerBlock = blockDim.x >> 5;
    const int wave = blockIdx.x * wavesPerBlock + (threadIdx.x >> 5);
    const int lane = threadIdx.x & 31;

    const int b  = wave / QTILES;
    const int mi = wave % QTILES;    // query tile index
    const int m0 = mi * 16;
    const int col = lane & 15;
    const bool hiHalf = lane >= 16;

    const float* pb = p + (size_t)b * NPTS * 6;

    // Each lane loads query point m0 + (lane&15); 24B stride, b96 load.
    const float* q = pb + (size_t)(m0 + col) * 6;
    const float qx = q[0], qy = q[1], qz = q[2];
    const float qn = qx * qx + qy * qy + qz * qz;   // exactly the qn this lane's column needs

    // B-matrix 4x16 f32: lanes 0-15 carry K=0,1 ; lanes 16-31 carry K=2,3.
    v2f bq;
    bq.x = hiHalf ? (-2.0f * qz) : (-2.0f * qx);
    bq.y = hiHalf ? 1.0f         : (-2.0f * qy);

    int res = 0;
    bool found = false;

    // Diagonal (j == query index) lies in tile mi, so nj <= mi always suffices.
    for (int nj = 0; nj <= mi; ++nj) {
        const float* s = pb + (size_t)(nj * 16 + col) * 6;
        if (nj < mi)   // wave-uniform; prefetch next source tile (global_prefetch_b8)
            __builtin_prefetch(pb + (size_t)((nj + 1) * 16 + col) * 6, 0, 1);
        const float sx = s[0], sy = s[1], sz = s[2];
        const float sn = sx * sx + sy * sy + sz * sz;

        // A-matrix 16x4 f32: lanes 0-15 carry K=0,1 ; lanes 16-31 carry K=2,3.
        v2f as;
        as.x = hiHalf ? sz : sx;
        as.y = hiHalf ? sn : sy;

        v8f c = {};
        c = __builtin_amdgcn_wmma_f32_16x16x4_f32(
                /*neg_a=*/false, as, /*neg_b=*/false, bq,
                /*c_mod=*/(short)0, c, /*reuse_a=*/false, /*reuse_b=*/false);

        // Pack per-lane 8-bit match mask over this lane's 8 source rows.
        unsigned m8 = 0u;
#pragma unroll
        for (int r = 0; r < 8; ++r) {
            const float d2 = qn + c[r];
            m8 |= (d2 < 1.0f) ? (1u << r) : 0u;
        }

        // Merge halves: lanes<16 hold sources 0..7, lanes>=16 hold 8..15.
        const unsigned partner = (unsigned)__shfl_xor((int)m8, 16, 32);
        const unsigned mask16 = hiHalf ? (partner | (m8 << 8))
                                       : (m8 | (partner << 8));

        if (!found && mask16) {
            res = nj * 16 + __builtin_ctz(mask16);
            found = true;
        }

        const unsigned fm = (unsigned)__ballot(found);
        if (fm == 0xFFFFFFFFu) break;   // wave-uniform early exit
    }

    if (!hiHalf) out[(size_t)b * NPTS + m0 + lane] = res;
}

extern "C" void kernel_launch(void* const* d_in, const int* in_sizes, int n_in,
                              void* d_out, int out_size, void* d_ws, size_t ws_size,
                              hipStream_t stream) {
    (void)in_sizes; (void)n_in; (void)out_size; (void)d_ws; (void)ws_size;
    const float* p = (const float*)d_in[0];
    int* out = (int*)d_out;

    const int totalWaves  = NBATCH * QTILES;            // 2048 waves
    const int threads     = 256;                        // 8 waves / block
    const int blocks      = totalWaves * 32 / threads;  // 256 blocks
    hipLaunchKernelGGL(ball_query_wmma_kernel, dim3(blocks), dim3(threads), 0, stream,
                       p, out);
}